// GNN_90142773608680
// MI455X (gfx1250) — compile-verified
//
#include <hip/hip_runtime.h>
#include <hip/hip_bf16.h>

typedef __attribute__((ext_vector_type(2))) float v2f;
typedef __attribute__((ext_vector_type(8))) float v8f;

#define NNODES 50000
#define NEDGES 800000
#define IN_C 128
#define HID_C 128
#define OUT_C 64
#define KC 32  // K-chunk staged in LDS per iteration

// ---------------------------------------------------------------------------
// FP32 WMMA GEMM with LDS staging: C[M x N] = A[M x K] * B[K x N], row-major.
// M % 16 == 0, K % KC == 0, N % 16 == 0, N <= 128.
// Block = 256 threads (8 waves). Block computes a 16-row x N strip:
//   - all threads cooperatively stage A-tile chunk (16 x KC) and B chunk
//     (KC x N, contiguous span of B) into LDS
//   - wave w computes column tile [16w, 16w+16) via v_wmma_f32_16x16x4_f32
// Wave-uniform guard (col0 < N) keeps EXEC all-ones inside WMMA waves.
// ---------------------------------------------------------------------------
__global__ __launch_bounds__(256) void wmma_gemm_f32(
    const float* __restrict__ A, const float* __restrict__ B,
    float* __restrict__ C, int M, int K, int N) {
  __shared__ float sA[16 * KC];    // 2 KB
  __shared__ float sB[KC * 128];   // 16 KB max

  const int tid  = threadIdx.x;
  const int lane = tid & 31;
  const int wave = tid >> 5;
  const int row0 = blockIdx.x << 4;  // M tile base
  const int col0 = wave << 4;        // N tile base (may exceed N for small N)
  const int r    = lane & 15;
  const int kk   = (lane >> 4) << 1; // lanes 0-15 -> K {0,1}; 16-31 -> K {2,3}
  const bool compute = (col0 < N);   // wave-uniform

  v8f acc = {};

  for (int kb = 0; kb < K; kb += KC) {
    __syncthreads();  // previous chunk's LDS reads complete before overwrite

    // Stage A chunk: rows row0..row0+15, cols kb..kb+KC-1 (strided rows)
    for (int i = tid; i < 16 * KC; i += 256) {
      int ar = i >> 5;          // KC == 32
      int ac = i & (KC - 1);
      sA[i] = A[(size_t)(row0 + ar) * K + kb + ac];
    }
    // Stage B chunk: contiguous span B[kb*N .. kb*N + KC*N), float4 copies
    {
      const float4* Bsrc = reinterpret_cast<const float4*>(B + (size_t)kb * N);
      float4* Bdst = reinterpret_cast<float4*>(sB);
      const int nv = (KC * N) >> 2;
      for (int i = tid; i < nv; i += 256) Bdst[i] = Bsrc[i];
    }
    __syncthreads();

    if (compute) {
#pragma unroll
      for (int k = 0; k < KC; k += 4) {
        v2f a, b;
        a.x = sA[r * KC + kk + k];
        a.y = sA[r * KC + kk + k + 1];
        b.x = sB[(kk + k) * N + col0 + r];
        b.y = sB[(kk + k + 1) * N + col0 + r];
        // D = A*B + C ; (neg_a, A, neg_b, B, c_mod, C, reuse_a, reuse_b)
        acc = __builtin_amdgcn_wmma_f32_16x16x4_f32(
            false, a, false, b, (short)0, acc, false, false);
      }
    }
  }

  if (compute) {
    // C/D layout: VGPR i -> M=row0+i (lanes 0-15) / row0+8+i (lanes 16-31); N=col0+r
    const int mBase = row0 + ((lane >> 4) << 3);
    float* Cout = C + (size_t)mBase * N + (col0 + r);
#pragma unroll
    for (int i = 0; i < 8; ++i) Cout[(size_t)i * N] = acc[i];
  }
}

// ---------------------------------------------------------------------------
// Degree / normalization
// ---------------------------------------------------------------------------
__global__ void k_init_deg(float* deg, int n) {
  int i = blockIdx.x * blockDim.x + threadIdx.x;
  if (i < n) deg[i] = 1.0f;  // self-loop
}

__global__ void k_deg_edges(const int* __restrict__ dst, float* deg, int e) {
  int i = blockIdx.x * blockDim.x + threadIdx.x;
  if (i < e) atomicAdd(&deg[dst[i]], 1.0f);
}

__global__ void k_dinv(float* deg, int n) {  // in place: deg -> rsqrt(deg)
  int i = blockIdx.x * blockDim.x + threadIdx.x;
  if (i < n) deg[i] = __frsqrt_rn(deg[i]);
}

// ---------------------------------------------------------------------------
// Self-loop contribution: out[i][c] = H[i][c] * dinv[i]^2 (+ bias)
// cshift = log2(C) -> no integer division
// ---------------------------------------------------------------------------
__global__ void k_self_init(const float* __restrict__ H,
                            const float* __restrict__ dinv,
                            const float* __restrict__ bias,  // may be null
                            float* __restrict__ out, int n, int cshift) {
  long long i = (long long)blockIdx.x * blockDim.x + threadIdx.x;
  long long total = (long long)n << cshift;
  if (i >= total) return;
  int row = (int)(i >> cshift);
  int ch  = (int)(i & ((1 << cshift) - 1));
  float dv = dinv[row];
  float v = H[i] * dv * dv;
  if (bias) v += bias[ch];
  out[i] = v;
}

// ---------------------------------------------------------------------------
// Edge scatter: out[dst] += H[src] * dinv[src]*dinv[dst]; float4 channel groups
// qshift = log2(C/4); one thread per (edge, 4-channel group)
// ---------------------------------------------------------------------------
__global__ void k_edge_scatter(const float* __restrict__ H,
                               float* __restrict__ out,
                               const int* __restrict__ src,
                               const int* __restrict__ dst,
                               const float* __restrict__ dinv,
                               int e, int c, int qshift) {
  long long gid = (long long)blockIdx.x * blockDim.x + threadIdx.x;
  long long total = (long long)e << qshift;
  if (gid >= total) return;
  int ei = (int)(gid >> qshift);
  int q  = (int)(gid & ((1 << qshift) - 1));
  int s = src[ei], d = dst[ei];
  float nrm = dinv[s] * dinv[d];
  const float4 h = reinterpret_cast<const float4*>(H + (size_t)s * c)[q];
  float* o = out + (size_t)d * c + 4 * q;
  atomicAdd(o + 0, h.x * nrm);
  atomicAdd(o + 1, h.y * nrm);
  atomicAdd(o + 2, h.z * nrm);
  atomicAdd(o + 3, h.w * nrm);
}

// ---------------------------------------------------------------------------
// H2 = relu(A1 + b1), in place OK; cshift = log2(C)
// ---------------------------------------------------------------------------
__global__ void k_bias_relu(float* __restrict__ a, const float* __restrict__ b,
                            int n, int cshift) {
  long long i = (long long)blockIdx.x * blockDim.x + threadIdx.x;
  long long total = (long long)n << cshift;
  if (i >= total) return;
  int ch = (int)(i & ((1 << cshift) - 1));
  float v = a[i] + b[ch];
  a[i] = v > 0.0f ? v : 0.0f;
}

static inline int cdiv(long long a, int b) { return (int)((a + b - 1) / b); }

extern "C" void kernel_launch(void* const* d_in, const int* in_sizes, int n_in,
                              void* d_out, int out_size, void* d_ws, size_t ws_size,
                              hipStream_t stream) {
  const float* x   = (const float*)d_in[0];          // [N, 128]
  const int*   ei  = (const int*)d_in[1];            // [2, E] row-major
  const float* W1  = (const float*)d_in[2];          // [128, 128]
  const float* b1  = (const float*)d_in[3];          // [128]
  const float* W2  = (const float*)d_in[4];          // [128, 64]
  const float* b2  = (const float*)d_in[5];          // [64]
  float*       out = (float*)d_out;                  // [N, 64]

  const int* src = ei;
  const int* dst = ei + NEDGES;

  // workspace layout (floats): dinv | H1 | A1 | H3  (~64.3 MB total)
  float* ws   = (float*)d_ws;
  float* dinv = ws;                                  // N
  float* H1   = dinv + NNODES;                       // N*128
  float* A1   = H1 + (size_t)NNODES * HID_C;         // N*128 (becomes H2 after relu)
  float* H3   = A1 + (size_t)NNODES * HID_C;         // N*64

  const int T = 256;

  // 1) degrees + normalization
  k_init_deg<<<cdiv(NNODES, T), T, 0, stream>>>(dinv, NNODES);
  k_deg_edges<<<cdiv(NEDGES, T), T, 0, stream>>>(dst, dinv, NEDGES);
  k_dinv<<<cdiv(NNODES, T), T, 0, stream>>>(dinv, NNODES);

  // 2) H1 = X @ W1   (M=50000, K=128, N=128)
  wmma_gemm_f32<<<NNODES / 16, 256, 0, stream>>>(x, W1, H1, NNODES, IN_C, HID_C);

  // 3) A1 = self-loop + edge aggregation of H1
  k_self_init<<<cdiv((long long)NNODES * HID_C, T), T, 0, stream>>>(
      H1, dinv, nullptr, A1, NNODES, 7 /* log2(128) */);
  k_edge_scatter<<<cdiv((long long)NEDGES * (HID_C / 4), T), T, 0, stream>>>(
      H1, A1, src, dst, dinv, NEDGES, HID_C, 5 /* log2(128/4) */);

  // 4) H2 = relu(A1 + b1), in place
  k_bias_relu<<<cdiv((long long)NNODES * HID_C, T), T, 0, stream>>>(
      A1, b1, NNODES, 7);

  // 5) H3 = H2 @ W2   (M=50000, K=128, N=64)
  wmma_gemm_f32<<<NNODES / 16, 256, 0, stream>>>(A1, W2, H3, NNODES, HID_C, OUT_C);

  // 6) out = self-loop*H3 + b2, then edge aggregation
  k_self_init<<<cdiv((long long)NNODES * OUT_C, T), T, 0, stream>>>(
      H3, dinv, b2, out, NNODES, 6 /* log2(64) */);
  k_edge_scatter<<<cdiv((long long)NEDGES * (OUT_C / 4), T), T, 0, stream>>>(
      H3, out, src, dst, dinv, NEDGES, OUT_C, 4 /* log2(64/4) */);
}